// TreeAggregatorCell_18717467476619
// MI455X (gfx1250) — compile-verified
//
#include <hip/hip_runtime.h>
#include <hip/hip_bf16.h>

typedef float v2f __attribute__((ext_vector_type(2)));
typedef float v8f __attribute__((ext_vector_type(8)));
typedef int   v4i __attribute__((vector_size(16)));   // matches builtin param type

#if defined(__has_builtin)
#  if __has_builtin(__builtin_amdgcn_global_load_async_to_lds_b128) && \
      __has_builtin(__builtin_amdgcn_s_wait_asynccnt)
#    define HAVE_ASYNC_LDS 1
#  endif
#endif
#ifndef HAVE_ASYNC_LDS
#  define HAVE_ASYNC_LDS 0
#endif

__device__ __forceinline__ v8f wmma4(v2f a, v2f b, v8f c) {
#if defined(__HIP_DEVICE_COMPILE__)
    // D = A(16x4,f32) * B(4x16,f32) + C(16x16,f32)
    return __builtin_amdgcn_wmma_f32_16x16x4_f32(false, a, false, b, (short)0, c,
                                                 false, false);
#else
    (void)a; (void)b;
    return c;
#endif
}

__device__ __forceinline__ float sigmoidf_fast(float v) {
    return 1.0f / (1.0f + __expf(-v));
}

// ---------------------------------------------------------------------------
// Pass A: xproj[n, 0:128]   = x @ W_f.T   + b_f
//         xproj[n, 128:256] = x @ W_att.T + b_att
//         xproj[n, 256:640] = x @ W_iou.T + b_iou
// grid (N/16, 5), block 256 (8 waves); wave handles one 16x16 output tile.
// ---------------------------------------------------------------------------
__global__ __launch_bounds__(256) void xproj_kernel(
    const float* __restrict__ x,
    const float* __restrict__ W_att, const float* __restrict__ b_att,
    const float* __restrict__ W_iou, const float* __restrict__ b_iou,
    const float* __restrict__ W_f,   const float* __restrict__ b_f,
    float* __restrict__ xproj)
{
    const int lane = threadIdx.x & 31;
    const int wave = threadIdx.x >> 5;
    const int n0 = blockIdx.x * 16;                 // node tile base
    const int ct = blockIdx.y * 8 + wave;           // column tile 0..39
    const int g0 = ct * 16;                         // column base in [0,640)

    const float* W; const float* bias; int grow;
    if (g0 < 128)      { W = W_f;   bias = b_f;   grow = g0;       }
    else if (g0 < 256) { W = W_att; bias = b_att; grow = g0 - 128; }
    else               { W = W_iou; bias = b_iou; grow = g0 - 256; }

    const int m  = lane & 15;            // A-row / B-col / D-col within tile
    const int kb = (lane >> 4) * 2;      // K sub-offset (0 or 2)
    const float* xrow = x + (size_t)(n0 + m) * 128;
    const float* wrow = W + (size_t)(grow + m) * 128;

    v8f acc = {};
    for (int k0 = 0; k0 < 128; k0 += 4) {
        v2f a, b;
        a.x = xrow[k0 + kb]; a.y = xrow[k0 + kb + 1];
        b.x = wrow[k0 + kb]; b.y = wrow[k0 + kb + 1];
        acc = wmma4(a, b, acc);
    }
    const float bv = bias[grow + m];
    const int rofs = (lane >> 4) * 8;    // D rows: j (lanes 0-15) / j+8 (16-31)
    for (int j = 0; j < 8; ++j) {
        int row = n0 + j + rofs;
        xproj[(size_t)row * 640 + g0 + m] = acc[j] + bv;
    }
}

// ---------------------------------------------------------------------------
// Pass B: per-node mailbox processing. 1 block = 1 node, 256 threads (8 waves).
//   F  = mh @ U_f.T   -> f = sigmoid(F + xWf), c_agg = sum_k f*mc
//   Wh = mh @ W_att.T -> scores -> softmax -> h_tilde = sum_k w_k * mh[k]
// ---------------------------------------------------------------------------
#define MH_STRIDE 132   // 132 % 64 == 4 -> conflict-free WMMA A-fragment reads

__global__ __launch_bounds__(256) void node_kernel(
    const float* __restrict__ mailbox_h, const float* __restrict__ mailbox_c,
    const float* __restrict__ tvec,
    const float* __restrict__ U_f, const float* __restrict__ W_att,
    const float* __restrict__ b_att,
    const float* __restrict__ u_w, const float* __restrict__ u_b,
    const float* __restrict__ xproj,
    float* __restrict__ c_agg_out, float* __restrict__ h_tilde_out)
{
    __shared__ __align__(16) float mh[32 * MH_STRIDE];
    __shared__ __align__(16) float mc[32 * 128];
    __shared__ __align__(16) float Fg[32 * MH_STRIDE];
    __shared__ __align__(16) float Wh[32 * MH_STRIDE];
    __shared__ float red[2 * 128];
    __shared__ float sp[32 * 8];
    __shared__ float wxp[8], cbp[8];
    __shared__ float sc[32], wts[32];

    const int n    = blockIdx.x;
    const int tid  = threadIdx.x;
    const int lane = tid & 31;
    const int wave = tid >> 5;

    // 1. stage mh (+ per-edge time shift) and mc into LDS.
    //    mh: b128 loads, broadcast-add time[k], b128 LDS stores.
    //    mc: pure copy -> async global->LDS b128 (ASYNCcnt path) when available.
    const float4* mh4  = (const float4*)(mailbox_h + (size_t)n * 4096);
    const float4* mc4  = (const float4*)(mailbox_c + (size_t)n * 4096);
    const float*  tsrc = tvec + (size_t)n * 32;

#if HAVE_ASYNC_LDS && defined(__HIP_DEVICE_COMPILE__)
    for (int i = tid; i < 1024; i += 256) {
        __builtin_amdgcn_global_load_async_to_lds_b128(
            (__attribute__((address_space(1))) v4i*)(mc4 + i),
            (__attribute__((address_space(3))) v4i*)&mc[i * 4],
            0, 0);
    }
#endif
    for (int i = tid; i < 1024; i += 256) {        // 32 rows x 32 quads
        const int k = i >> 5, h4 = (i & 31) * 4;
        float4 v = mh4[i];
        const float tk = tsrc[k];
        v.x += tk; v.y += tk; v.z += tk; v.w += tk;
        *(float4*)&mh[k * MH_STRIDE + h4] = v;
#if !(HAVE_ASYNC_LDS && defined(__HIP_DEVICE_COMPILE__))
        *(float4*)&mc[i * 4] = mc4[i];
#endif
    }
#if HAVE_ASYNC_LDS && defined(__HIP_DEVICE_COMPILE__)
    __builtin_amdgcn_s_wait_asynccnt(0);
#endif
    __syncthreads();

    // 2. WMMA: wave w produces cols [16w,16w+16) of F and Wh (rows 0..31)
    {
        const int m  = lane & 15;
        const int kb = (lane >> 4) * 2;
        const int g  = wave * 16 + m;                 // output column
        const float* ufrow = U_f   + (size_t)g * 128;
        const float* warow = W_att + (size_t)g * 128;
        const float* a0p = &mh[(size_t)m * MH_STRIDE];         // rows 0..15
        const float* a1p = &mh[(size_t)(16 + m) * MH_STRIDE];  // rows 16..31
        v8f accF0 = {}, accF1 = {}, accW0 = {}, accW1 = {};
        for (int k0 = 0; k0 < 128; k0 += 4) {
            v2f a0, a1, b1, b2;
            a0.x = a0p[k0 + kb];   a0.y = a0p[k0 + kb + 1];
            a1.x = a1p[k0 + kb];   a1.y = a1p[k0 + kb + 1];
            b1.x = ufrow[k0 + kb]; b1.y = ufrow[k0 + kb + 1];
            b2.x = warow[k0 + kb]; b2.y = warow[k0 + kb + 1];
            accF0 = wmma4(a0, b1, accF0);
            accF1 = wmma4(a1, b1, accF1);
            accW0 = wmma4(a0, b2, accW0);
            accW1 = wmma4(a1, b2, accW1);
        }
        const int rofs = (lane >> 4) * 8;
        const int col  = wave * 16 + m;
        for (int j = 0; j < 8; ++j) {
            Fg[(j + rofs) * MH_STRIDE + col]      = accF0[j];
            Fg[(16 + j + rofs) * MH_STRIDE + col] = accF1[j];
            Wh[(j + rofs) * MH_STRIDE + col]      = accW0[j];
            Wh[(16 + j + rofs) * MH_STRIDE + col] = accW1[j];
        }
    }
    __syncthreads();

    // 3a. c_agg partials: f = sigmoid(F + xWf + b_f), sum_k f * mc
    {
        const int g = tid & 127, half = tid >> 7;
        const float xwf = xproj[(size_t)n * 640 + g];   // includes b_f
        float acc = 0.f;
        for (int k = half * 16; k < half * 16 + 16; ++k) {
            float fv = sigmoidf_fast(Fg[k * MH_STRIDE + g] + xwf);
            acc += fv * mc[k * 128 + g];
        }
        red[half * 128 + g] = acc;
    }
    // 3b. score partials: sum_g Wh[k,g]*u_w[g]  (+ wx and b_att·u1 constants)
    {
        const int row = tid >> 3, seg = tid & 7;
        float p = 0.f;
        for (int c = seg * 16; c < seg * 16 + 16; ++c)
            p += Wh[row * MH_STRIDE + c] * u_w[c];
        sp[row * 8 + seg] = p;
        if (tid < 8) {
            float q = 0.f, cb = 0.f;
            for (int c = tid * 16; c < tid * 16 + 16; ++c) {
                q  += xproj[(size_t)n * 640 + 128 + c] * u_w[128 + c]; // Wx·u2
                cb += b_att[c] * u_w[c];                               // b_att·u1
            }
            wxp[tid] = q; cbp[tid] = cb;
        }
    }
    __syncthreads();

    if (tid < 128)
        c_agg_out[(size_t)n * 128 + tid] = red[tid] + red[128 + tid];

    if (tid < 32) {   // finish scores + leaky_relu (slope 0.01)
        float s = 0.f;
        for (int j = 0; j < 8; ++j) s += sp[tid * 8 + j];
        float wx = 0.f, cb = 0.f;
        for (int j = 0; j < 8; ++j) { wx += wxp[j]; cb += cbp[j]; }
        s += wx + cb + u_b[0];
        sc[tid] = (s > 0.f) ? s : 0.01f * s;
    }
    __syncthreads();

    if (tid < 32) {   // softmax over 32 children (redundant per-thread scan)
        float mx = -1e30f;
        for (int j = 0; j < 32; ++j) mx = fmaxf(mx, sc[j]);
        float sum = 0.f;
        for (int j = 0; j < 32; ++j) sum += __expf(sc[j] - mx);
        wts[tid] = __expf(sc[tid] - mx) / sum;
    }
    __syncthreads();

    {   // h_tilde = sum_k wts[k] * mh[k,:]
        const int g = tid & 127, half = tid >> 7;
        float acc = 0.f;
        for (int k = half * 16; k < half * 16 + 16; ++k)
            acc += wts[k] * mh[k * MH_STRIDE + g];
        red[half * 128 + g] = acc;
    }
    __syncthreads();
    if (tid < 128)
        h_tilde_out[(size_t)n * 128 + tid] = red[tid] + red[128 + tid];
}

// ---------------------------------------------------------------------------
// Pass C: iou = h_tilde @ U_iou.T + xWiou(+b_iou);  c/h epilogue.
// grid N/16, block 256 (8 waves x 3 column tiles = 24 tiles of 16 cols).
// ---------------------------------------------------------------------------
#define IOU_STRIDE 392   // 392 % 64 == 8, multiple of 4 for b128 access

__global__ __launch_bounds__(256) void iou_kernel(
    const float* __restrict__ U_iou, const float* __restrict__ xproj,
    const float* __restrict__ h_tilde_ws, const float* __restrict__ c_agg_ws,
    float* __restrict__ out_h, float* __restrict__ out_c)
{
    __shared__ __align__(16) float iou[16 * IOU_STRIDE];
    const int tid = threadIdx.x, lane = tid & 31, wave = tid >> 5;
    const int n0 = blockIdx.x * 16;
    const int m = lane & 15, kb = (lane >> 4) * 2, rofs = (lane >> 4) * 8;

    for (int t = 0; t < 3; ++t) {
        const int g0 = (wave * 3 + t) * 16;                // [0,384)
        const float* urow = U_iou + (size_t)(g0 + m) * 128;
        const float* arow = h_tilde_ws + (size_t)(n0 + m) * 128;
        v8f acc = {};
        for (int k0 = 0; k0 < 128; k0 += 4) {
            v2f a, b;
            a.x = arow[k0 + kb]; a.y = arow[k0 + kb + 1];
            b.x = urow[k0 + kb]; b.y = urow[k0 + kb + 1];
            acc = wmma4(a, b, acc);
        }
        for (int j = 0; j < 8; ++j) {
            int row = j + rofs;
            iou[row * IOU_STRIDE + g0 + m] =
                acc[j] + xproj[(size_t)(n0 + row) * 640 + 256 + g0 + m];
        }
    }
    __syncthreads();

    // epilogue: 16 rows x 128 cols as float4 (512 quads, 2 per thread)
    for (int idx = tid; idx < 512; idx += 256) {
        const int row = idx >> 5, g4 = (idx & 31) * 4;
        const float4 iv = *(const float4*)&iou[row * IOU_STRIDE + g4];
        const float4 ov = *(const float4*)&iou[row * IOU_STRIDE + 128 + g4];
        const float4 uv = *(const float4*)&iou[row * IOU_STRIDE + 256 + g4];
        const float4 cg = *(const float4*)&c_agg_ws[(size_t)(n0 + row) * 128 + g4];
        float4 c, h;
        c.x = sigmoidf_fast(iv.x) * tanhf(uv.x) + cg.x;
        c.y = sigmoidf_fast(iv.y) * tanhf(uv.y) + cg.y;
        c.z = sigmoidf_fast(iv.z) * tanhf(uv.z) + cg.z;
        c.w = sigmoidf_fast(iv.w) * tanhf(uv.w) + cg.w;
        h.x = sigmoidf_fast(ov.x) * tanhf(c.x);
        h.y = sigmoidf_fast(ov.y) * tanhf(c.y);
        h.z = sigmoidf_fast(ov.z) * tanhf(c.z);
        h.w = sigmoidf_fast(ov.w) * tanhf(c.w);
        *(float4*)&out_h[(size_t)(n0 + row) * 128 + g4] = h;
        *(float4*)&out_c[(size_t)(n0 + row) * 128 + g4] = c;
    }
}

// ---------------------------------------------------------------------------
extern "C" void kernel_launch(void* const* d_in, const int* in_sizes, int n_in,
                              void* d_out, int out_size, void* d_ws, size_t ws_size,
                              hipStream_t stream)
{
    const float* x     = (const float*)d_in[0];
    const float* mbh   = (const float*)d_in[1];
    const float* mbc   = (const float*)d_in[2];
    const float* tvec  = (const float*)d_in[3];
    const float* W_att = (const float*)d_in[4];
    const float* b_att = (const float*)d_in[5];
    const float* u_w   = (const float*)d_in[6];
    const float* u_b   = (const float*)d_in[7];
    const float* W_iou = (const float*)d_in[8];
    const float* U_iou = (const float*)d_in[9];
    const float* b_iou = (const float*)d_in[10];
    const float* U_f   = (const float*)d_in[11];
    const float* W_f   = (const float*)d_in[12];
    const float* b_f   = (const float*)d_in[13];

    const int N = in_sizes[0] / 128;   // XS = 128

    float* xproj = (float*)d_ws;                    // N*640
    float* c_agg = xproj + (size_t)N * 640;         // N*128
    float* h_til = c_agg + (size_t)N * 128;         // N*128
    float* out_h = (float*)d_out;                   // N*128
    float* out_c = out_h + (size_t)N * 128;         // N*128

    dim3 gA(N / 16, 5);
    xproj_kernel<<<gA, 256, 0, stream>>>(x, W_att, b_att, W_iou, b_iou,
                                         W_f, b_f, xproj);
    node_kernel<<<N, 256, 0, stream>>>(mbh, mbc, tvec, U_f, W_att, b_att,
                                       u_w, u_b, xproj, c_agg, h_til);
    iou_kernel<<<N / 16, 256, 0, stream>>>(U_iou, xproj, h_til, c_agg,
                                           out_h, out_c);
}